// YoloDetector_33612414058580
// MI455X (gfx1250) — compile-verified
//
#include <hip/hip_runtime.h>
#include <math.h>

// ---------------- problem constants (from reference) ----------------
#define Sg   7
#define Bb   2
#define Cc   20
#define NB   98      // S*S*B boxes
#define NN   1960    // S*S*B*C flattened detections
#define NPAD 2048    // padded for bitonic sort
#define SCo  980     // S*S*C  (class_prob section length)
#define Lp   1470    // total prediction length per image
#define NBATCH 16
#define Wimg 448.0f
#define IOU_THR  0.1f
#define PROB_THR 8e-25f

typedef float v2f __attribute__((ext_vector_type(2)));
typedef float v8f __attribute__((ext_vector_type(8)));

#if defined(__AMDGCN__) && __has_builtin(__builtin_amdgcn_global_load_async_to_lds_b32)
#define HAS_ASYNC 1
#else
#define HAS_ASYNC 0
#endif

#if defined(__AMDGCN__) && __has_builtin(__builtin_amdgcn_wmma_f32_16x16x4_f32)
#define HAS_WMMA4 1
#else
#define HAS_WMMA4 0
#endif

#if defined(__AMDGCN__)
typedef __attribute__((address_space(1))) int* glob_i32p;
typedef __attribute__((address_space(3))) int* lds_i32p;
#endif

// One workgroup (8 wave32s) per image. Entire pipeline is LDS-resident.
__global__ __launch_bounds__(256) void yolo_nms_kernel(const float* __restrict__ pred_all,
                                                       float* __restrict__ out,
                                                       int nbatch) {
    __shared__ float pred[Lp];        //  5.9 KB raw prediction
    __shared__ float eprob[NPAD];     //  8   KB class*conf products (flat n = m*20+c)
    __shared__ float key[NPAD];       //  8   KB sort keys -> suppressed probs
    __shared__ int   sidx[NPAD];      //  8   KB sort payload (original flat index)
    __shared__ float bx[NB], by[NB], bw[NB], bh[NB]; // decoded boxes
    __shared__ int   clsb[NB];        // per-box argmax class
    // total ~32.4 KB  (WGP LDS = 320 KB)

    const int tid   = threadIdx.x;
    const int batch = blockIdx.x;
    if (batch >= nbatch) return;
    const float* pg = pred_all + (size_t)batch * Lp;

    // ---------- Phase 0: stage prediction into LDS (async global->LDS) ----------
#if HAS_ASYNC
    for (int i = tid; i < Lp; i += 256) {
        __builtin_amdgcn_global_load_async_to_lds_b32(
            (glob_i32p)(pg + i),
            (lds_i32p)(&pred[i]),
            /*offset=*/0, /*cpol=*/0);
    }
    asm volatile("s_wait_asynccnt 0" ::: "memory");
#else
    for (int i = tid; i < Lp; i += 256) pred[i] = pg[i];
#endif
    __syncthreads();

    // ---------- Phase 1a: decode boxes (threads 0..97) ----------
    if (tid < NB) {
        const int m  = tid;
        const int s  = m / (Sg * Bb);        // grid row
        const int s2 = (m / Bb) % Sg;        // grid col
        const float* braw = &pred[SCo + NB + m * 4];
        bx[m] = (braw[0] + (float)s)  * (Wimg / (float)Sg);
        by[m] = (braw[1] + (float)s2) * (Wimg / (float)Sg);
        bw[m] = braw[2] * braw[2] * Wimg;
        bh[m] = braw[3] * braw[3] * Wimg;
    }

    // ---------- Phase 1b: e_probs = class_prob (x) confidence ----------
    // Rank-1 outer products per cell, batched 4 cells per V_WMMA_F32_16X16X4_F32:
    //   A[16x4]: row r(<8) -> cell 4t+r/2, box r%2; A[r][k] = (k==r/2)?conf:0
    //   B[4x16]: B[k][c]   = class_prob[cell 4t+k][c]   (two column halves for C=20)
    //   D[r][c] = conf * class_prob  (exact f32; f16 would underflow PROB_THR=8e-25)
#if HAS_WMMA4
    if (tid < 32) {                     // whole wave0 enters: EXEC all ones
        const int lane = tid;
        const int g  = lane >> 4;       // lane half selects K pair {0,1} vs {2,3}
        const int r  = lane & 15;       // A: M index / B,D: N index
        const int kA = g ? 2 : 0;
        const int kB = g ? 3 : 1;
        for (int t = 0; t < 13; ++t) {  // cells 4t..4t+3  (49 cells total)
            v2f a; a.x = 0.0f; a.y = 0.0f;
            if (r < 8) {
                const int cell = 4 * t + (r >> 1);
                const float cv = (cell < 49) ? pred[SCo + cell * 2 + (r & 1)] : 0.0f;
                const int kk = r >> 1;
                a.x = (kk == kA) ? cv : 0.0f;
                a.y = (kk == kB) ? cv : 0.0f;
            }
            for (int half = 0; half < 2; ++half) {
                const int col = (half ? 16 : 0) + r;   // class index this lane covers
                v2f b; b.x = 0.0f; b.y = 0.0f;
                if (col < Cc) {
                    const int cA = 4 * t + kA, cB = 4 * t + kB;
                    b.x = (cA < 49) ? pred[cA * Cc + col] : 0.0f;
                    b.y = (cB < 49) ? pred[cB * Cc + col] : 0.0f;
                }
                v8f c = {0.f, 0.f, 0.f, 0.f, 0.f, 0.f, 0.f, 0.f};
                v8f d = __builtin_amdgcn_wmma_f32_16x16x4_f32(
                    false, a, false, b, (short)0, c, false, false);
                // D layout: lanes 0..15 hold rows M=v (v=0..7) at column N=lane
                if (g == 0 && col < Cc) {
                    for (int v = 0; v < 8; ++v) {
                        const int boxid = 8 * t + v;   // cell*2 + box
                        if (boxid < NB) eprob[boxid * Cc + col] = d[v];
                    }
                }
            }
        }
    }
#else
    for (int n = tid; n < NN; n += 256) {
        const int m = n / Cc, c = n % Cc, cell = m / Bb;
        eprob[n] = pred[cell * Cc + c] * pred[SCo + m];
    }
#endif
    __syncthreads();

    // ---------- Phase 2: per-box argmax(mask) class, sort key/payload init ----------
    if (tid < NB) {
        int cl = 0;
        for (int c = Cc - 1; c >= 0; --c)
            if (eprob[tid * Cc + c] >= PROB_THR) cl = c;   // first True, else 0
        clsb[tid] = cl;
    }
    for (int n = tid; n < NPAD; n += 256) {
        key[n]  = (n < NN) ? eprob[n] : -1.0f;   // pad sinks below real probs (>=0)
        sidx[n] = n;
    }
    __syncthreads();

    // ---------- Phase 3: bitonic sort, descending, stable via index tiebreak ----------
    for (int k = 2; k <= NPAD; k <<= 1) {
        for (int j = k >> 1; j > 0; j >>= 1) {
            for (int i = tid; i < NPAD; i += 256) {
                const int ixj = i ^ j;
                if (ixj > i) {
                    const float ka = key[i], kb2 = key[ixj];
                    const int   ia = sidx[i], ib = sidx[ixj];
                    const bool greaterAB = (ka > kb2) || (ka == kb2 && ia < ib);
                    const bool wantDesc  = ((i & k) == 0);
                    if (greaterAB != wantDesc) {
                        key[i] = kb2; key[ixj] = ka;
                        sidx[i] = ib; sidx[ixj] = ia;
                    }
                }
            }
            __syncthreads();
        }
    }

    // ---------- Phase 4: greedy NMS scan (matches reference lax.scan exactly) ----------
    for (int i = 0; i < NN - 1; ++i) {
        const float p = key[i];          // uniform read -> no divergence at barrier
        if (p > 0.0f) {
            const int mp = sidx[i] / Cc;
            const float px = bx[mp], py = by[mp], pw = bw[mp], ph = bh[mp];
            const float parea = pw * ph;
            for (int jj = i + 1 + tid; jj < NN; jj += 256) {
                const int mc = sidx[jj] / Cc;
                const float cx = bx[mc], cy = by[mc], cw = bw[mc], ch = bh[mc];
                const float tb = fminf(px + 0.5f * pw, cx + 0.5f * cw)
                               - fmaxf(px - 0.5f * pw, cx - 0.5f * cw);
                const float lr = fminf(py + 0.5f * ph, cy + 0.5f * ch)
                               - fmaxf(py - 0.5f * ph, cy - 0.5f * ch);
                const float inter = (tb < 0.0f || lr < 0.0f) ? 0.0f : tb * lr;
                const float iou   = inter / (parea + cw * ch - inter);
                if (iou > IOU_THR) key[jj] = 0.0f;
            }
        }
        __syncthreads();
    }

    // ---------- Phase 5: write outputs (boxes_c, probs_nms, cls_c) ----------
    float* outB = out + (size_t)batch * NN * 4;
    float* outP = out + (size_t)nbatch * NN * 4 + (size_t)batch * NN;
    float* outC = out + (size_t)nbatch * NN * 4 + (size_t)nbatch * NN + (size_t)batch * NN;
    for (int jj = tid; jj < NN; jj += 256) {
        const int m = sidx[jj] / Cc;
        outB[jj * 4 + 0] = bx[m];
        outB[jj * 4 + 1] = by[m];
        outB[jj * 4 + 2] = bw[m];
        outB[jj * 4 + 3] = bh[m];
        outP[jj] = key[jj];
        outC[jj] = (float)clsb[m];
    }
}

extern "C" void kernel_launch(void* const* d_in, const int* in_sizes, int n_in,
                              void* d_out, int out_size, void* d_ws, size_t ws_size,
                              hipStream_t stream) {
    (void)n_in; (void)out_size; (void)d_ws; (void)ws_size;
    const float* pred = (const float*)d_in[0];
    float* out = (float*)d_out;
    const int nbatch = in_sizes[0] / Lp;   // = 16 for the reference config
    yolo_nms_kernel<<<dim3(nbatch), dim3(256), 0, stream>>>(pred, out, nbatch);
}